// Part_CAM_mask_67542655697208
// MI455X (gfx1250) — compile-verified
//
#include <hip/hip_runtime.h>

typedef float v2f __attribute__((ext_vector_type(2)));
typedef float v8f __attribute__((ext_vector_type(8)));

#define LCNT     12
#define BATCH    64
#define NDIM     201
#define DDIM     768
#define NWAVES   4
#define NTHREADS 128
#define TS       211   // padded LDS row stride (odd -> conflict-free column reads)

// Computes the 16-row block T = sub[11][0:16,:] @ sub[10] @ ... @ sub[0] with
// V_WMMA_F32_16X16X4_F32 (only row 0 is needed by the reference), then
// out[bb,p,:] = T[0,1:M] @ relu(features[bb, s:s+M-1, :]).
// M is a compile-time constant so tile loops fully unroll and the WMMA
// accumulators stay in statically-indexed VGPRs.
template<int M>
__device__ __forceinline__ void part_body(
    const float* __restrict__ x, const float* __restrict__ features,
    float* __restrict__ out, float* __restrict__ T,
    const int bb, const int p, const int e, const int s)
{
    constexpr int NT  = (M + 15) / 16;          // column tiles (13 or 7)
    constexpr int Mp  = NT * 16;                // padded width
    constexpr int TPW = (NT + NWAVES - 1) / NWAVES;  // tiles per wave (4 or 2)

    const int tid  = threadIdx.x;
    const int wave = tid >> 5;
    const int lane = tid & 31;
    const int lr   = lane & 15;                 // M/N index inside fragment
    const int lh   = lane >> 4;                 // K-half selector

    const long xplane = (long)NDIM * NDIM;

    // ---- init T = sub[L-1][0:16, 0:M], zero-padded to Mp columns ----
    {
        const float* xl = x + (long)((LCNT - 1) * BATCH + bb) * xplane;
        for (int i = tid; i < 16 * Mp; i += NTHREADS) {
            const int r = i / Mp, c = i - r * Mp;
            float v = 0.f;
            if (c < M) {
                const int ridx = (r == 0) ? e : (s + r - 1);
                const int cidx = (c == 0) ? e : (s + c - 1);
                v = xl[(long)ridx * NDIM + cidx];
            }
            T[r * TS + c] = v;
        }
    }

    // ---- per-tile gather column indices (independent of k: hoist) ----
    int  cidx_r[TPW];
    bool cv_r[TPW];
    #pragma unroll
    for (int ti = 0; ti < TPW; ++ti) {
        const int  t   = wave + ti * NWAVES;
        const int  cc  = t * 16 + lr;
        const bool cv  = (t < NT) && (cc < M);
        const int  ccl = cv ? cc : 0;           // clamp to a valid column
        cidx_r[ti] = (ccl == 0) ? e : (s + ccl - 1);
        cv_r[ti]   = cv;
    }

    // ---- chain: T = T @ sub[l], l = L-2 .. 0 ----
    for (int l = LCNT - 2; l >= 0; --l) {
        const float* xa = x + (long)(l * BATCH + bb) * xplane;
        __syncthreads();                        // T stable for reading

        v8f acc[TPW];
        #pragma unroll
        for (int ti = 0; ti < TPW; ++ti) acc[ti] = (v8f)0.0f;

        #pragma unroll 2
        for (int k0 = 0; k0 < M; k0 += 4) {
            const int ka = k0 + 2 * lh;         // this lane-half's K pair
            // A fragment: T[0:16, k0:k0+4] (shared by all tiles of this wave)
            v2f a;
            a.x = T[lr * TS + ka];
            a.y = T[lr * TS + ka + 1];          // padded cols are zero
            const bool va = (ka < M);
            const bool vb = (ka + 1 < M);
            const int  ra = va ? ((ka == 0) ? e : (s + ka - 1)) : 0;  // clamped
            const int  rb = vb ? (s + ka) : 0;                        // clamped
            const float* xra = xa + (long)ra * NDIM;
            const float* xrb = xa + (long)rb * NDIM;

            #pragma unroll
            for (int ti = 0; ti < TPW; ++ti) {
                // Branchless predicated gather: always-valid address, then select.
                const float bx = xra[cidx_r[ti]];
                const float by = xrb[cidx_r[ti]];
                v2f bf;
                bf.x = (cv_r[ti] && va) ? bx : 0.f;
                bf.y = (cv_r[ti] && vb) ? by : 0.f;
                // D(16x16,f32) = A(16x4,f32) x B(4x16,f32) + C
                acc[ti] = __builtin_amdgcn_wmma_f32_16x16x4_f32(
                    false, a, false, bf, (short)0, acc[ti], false, false);
            }
        }

        __syncthreads();                        // all reads of T done
        #pragma unroll
        for (int ti = 0; ti < TPW; ++ti) {
            const int t = wave + ti * NWAVES;
            if (t < NT) {                       // wave-uniform
                const int c0 = t * 16 + lr;
                #pragma unroll
                for (int i = 0; i < 8; ++i)     // C/D layout: M = i + 8*lh, N = lr
                    T[(i + 8 * lh) * TS + c0] = acc[ti][i];
            }
        }
    }
    __syncthreads();

    // ---- phase 2: out[bb, p, :] = T[0, 1:M] @ relu(features[bb, s:s+M-1, :]) ----
    float* orow = out + ((long)bb * 4 + p) * DDIM;
    const float* frow = features + ((long)bb * NDIM + s) * DDIM;
    for (int d = tid; d < DDIM; d += NTHREADS) {
        float accd = 0.f;
        for (int k = 0; k < M - 1; ++k) {
            const float wv = T[k + 1];                    // LDS broadcast
            const float fv = frow[(long)k * DDIM + d];    // coalesced
            accd += wv * (fv > 0.f ? fv : 0.f);
        }
        orow[d] = accd;
    }
}

__global__ __launch_bounds__(NTHREADS) void part_cam_kernel(
    const float* __restrict__ x, const float* __restrict__ features,
    float* __restrict__ out)
{
    __shared__ float T[16 * TS];

    const int bb = blockIdx.x;                  // batch
    const int p  = blockIdx.y;                  // 0..3 -> part e = p+1
    const int e  = p + 1;

    if (p == 0) {
        part_body<197>(x, features, out, T, bb, p, e, 5);
    } else {
        const int s = (p == 1) ? 5 : ((p == 2) ? 54 : 103);
        part_body<99>(x, features, out, T, bb, p, e, s);
    }
}

extern "C" void kernel_launch(void* const* d_in, const int* in_sizes, int n_in,
                              void* d_out, int out_size, void* d_ws, size_t ws_size,
                              hipStream_t stream) {
    (void)in_sizes; (void)n_in; (void)d_ws; (void)ws_size; (void)out_size;
    const float* x        = (const float*)d_in[0];   // (12, 64, 201, 201) f32
    const float* features = (const float*)d_in[1];   // (64, 201, 768)  f32
    float* out            = (float*)d_out;           // (64, 4, 768)    f32

    dim3 grid(BATCH, 4);
    part_cam_kernel<<<grid, NTHREADS, 0, stream>>>(x, features, out);
}